// HeavyPrompt_88965952569882
// MI455X (gfx1250) — compile-verified
//
#include <hip/hip_runtime.h>
#include <hip/hip_bf16.h>
#include <stdint.h>

// ---------------------------------------------------------------------------
// HeavyPrompt graph-prompt assembly for MI455X (gfx1250).
// Pure bandwidth problem (~470 MB traffic -> ~20us at 23.3 TB/s HBM).
// Bulk x/pe copies use the CDNA5 async global<->LDS DMA path
// (global_load_async_to_lds_b128 / global_store_async_from_lds_b128,
//  tracked with ASYNCcnt) so data never touches VGPRs.
// ---------------------------------------------------------------------------

// Problem constants (match the reference).
constexpr int T_TOK   = 10;
constexpr int N_NODES = 4096;
constexpr int E_G     = 65536;
constexpr int D_FEAT  = 128;   // 32 float4 per row
constexpr int PE_DIM  = 32;    // 8 float4 per row
constexpr int SPD_L   = 4;
constexpr int BATCH   = 64;

constexpr int TPN  = T_TOK + N_NODES;          // 4106 rows per batch
constexpr int EI   = T_TOK * (T_TOK - 1);      // 90 inner edges
constexpr int TN   = T_TOK * N_NODES;          // 40960
constexpr int EC   = 2 * TN;                   // 81920 cross edges
constexpr int ETOT = EI + E_G + EC;            // 147546 edges per batch
constexpr long NEDGE = (long)BATCH * ETOT;     // 9,442,944

// d_out element offsets (4-byte units)
constexpr long OFF_X    = 0;                       // 33,636,352 f32
constexpr long OFF_PE   = 33636352L;               //  8,409,088 f32
constexpr long OFF_SRC  = 42045440L;               //  9,442,944 i32
constexpr long OFF_DST  = OFF_SRC + NEDGE;         //  9,442,944 i32
constexpr long OFF_W    = 60931328L;               //  9,442,944 f32
constexpr long OFF_ROOT = 70374272L;               //         64 i32

// --------------------------- async copy helpers ----------------------------
__device__ __forceinline__ unsigned lds_off_u32(const void* p) {
  // Flat LDS addresses are {SHARED_BASE, offset}: low 32 bits ARE the
  // wave-relative LDS byte address the async ops expect in VDST/VSRC.
  return (unsigned)(uintptr_t)p;
}

__device__ __forceinline__ void async_load_b128(unsigned lds, const void* gaddr) {
  asm volatile("global_load_async_to_lds_b128 %0, %1, off"
               :: "v"(lds), "v"(gaddr) : "memory");
}

__device__ __forceinline__ void async_store_b128(void* gaddr, unsigned lds) {
  asm volatile("global_store_async_from_lds_b128 %0, %1, off"
               :: "v"(gaddr), "v"(lds) : "memory");
}

__device__ __forceinline__ void wait_async0() {
  asm volatile("s_wait_asynccnt 0" ::: "memory");
}

// ----------------------------- x assembly ----------------------------------
// out_x[b*4106 + r] = (r < T) ? token_list[r] : g_x[b, r-T], row = 128 floats.
// Grid sized exactly: 8,409,088 float4 = 8212 blocks * 256 thr * 4 unroll.
constexpr int XU = 4;

__global__ __launch_bounds__(256)
void x_copy_async(const float4* __restrict__ tok,
                  const float4* __restrict__ gx,
                  float4* __restrict__ out) {
  __shared__ float4 buf[256 * XU];
  const int tid = threadIdx.x;
  const long g0 = (long)blockIdx.x * (256 * XU) + tid;
  const unsigned lds0 = lds_off_u32(&buf[tid]);

#pragma unroll
  for (int u = 0; u < XU; ++u) {
    long g = g0 + (long)u * 256;
    unsigned row = (unsigned)(g >> 5);      // 32 float4 per row
    unsigned c   = (unsigned)g & 31u;
    unsigned b   = row / (unsigned)TPN;
    unsigned r   = row - b * (unsigned)TPN;
    const float4* src = (r < (unsigned)T_TOK)
        ? (tok + (size_t)r * 32u + c)
        : (gx + ((size_t)b * N_NODES + (r - T_TOK)) * 32u + c);
    async_load_b128(lds0 + u * 256 * 16, src);
  }
  wait_async0();                             // LDS slots populated
#pragma unroll
  for (int u = 0; u < XU; ++u) {
    long g = g0 + (long)u * 256;
    async_store_b128(out + g, lds0 + u * 256 * 16);
  }
  wait_async0();                             // drain before LDS dealloc
}

// ----------------------------- pe assembly ---------------------------------
// Same pattern, 8 float4 per row. 2,102,272 float4 = 2053 blocks * 1024.
__global__ __launch_bounds__(256)
void pe_copy_async(const float4* __restrict__ tokpe,
                   const float4* __restrict__ gpe,
                   float4* __restrict__ out) {
  __shared__ float4 buf[256 * XU];
  const int tid = threadIdx.x;
  const long g0 = (long)blockIdx.x * (256 * XU) + tid;
  const unsigned lds0 = lds_off_u32(&buf[tid]);

#pragma unroll
  for (int u = 0; u < XU; ++u) {
    long g = g0 + (long)u * 256;
    unsigned row = (unsigned)(g >> 3);      // 8 float4 per row
    unsigned c   = (unsigned)g & 7u;
    unsigned b   = row / (unsigned)TPN;
    unsigned r   = row - b * (unsigned)TPN;
    const float4* src = (r < (unsigned)T_TOK)
        ? (tokpe + (size_t)r * 8u + c)
        : (gpe + ((size_t)b * N_NODES + (r - T_TOK)) * 8u + c);
    async_load_b128(lds0 + u * 256 * 16, src);
  }
  wait_async0();
#pragma unroll
  for (int u = 0; u < XU; ++u) {
    long g = g0 + (long)u * 256;
    async_store_b128(out + g, lds0 + u * 256 * 16);
  }
  wait_async0();
}

// --------------------------- edges + weights -------------------------------
// One thread per (batch, edge). Edge layout per batch:
//   [0, EI)            : inner token-token mesh (i != j, row-major)
//   [EI, EI+E_G)       : graph edges  (g_edge_index[b] + T)
//   [EI+E_G, ETOT)     : cross edges fwd (t -> n+T) then reversed (n+T -> t)
__global__ __launch_bounds__(256)
void edges_kernel(const int* __restrict__ g_ei,
                  const int* __restrict__ g_spd,
                  const float* __restrict__ sew,
                  const float* __restrict__ ewp,
                  int* __restrict__ out_src,
                  int* __restrict__ out_dst,
                  float* __restrict__ out_w) {
  long idx = (long)blockIdx.x * blockDim.x + threadIdx.x;
  if (idx >= NEDGE) return;
  int b = (int)(idx / ETOT);
  int e = (int)(idx - (long)b * ETOT);
  int off = b * TPN;

  int s, d;
  float w;
  if (e < EI) {                       // inner prompt mesh
    int i = e / (T_TOK - 1);
    int k = e - i * (T_TOK - 1);
    int j = k + (k >= i ? 1 : 0);
    s = i; d = j; w = *sew;
  } else if (e < EI + E_G) {          // original graph edges
    int k = e - EI;
    const int* base = g_ei + (long)b * 2 * E_G;
    s = base[k] + T_TOK;
    d = base[E_G + k] + T_TOK;
    w = 1.0f;
  } else {                            // undirected cross edges
    int c = e - (EI + E_G);
    bool fwd = (c < TN);
    if (!fwd) c -= TN;
    int t = c >> 12;                  // /N_NODES
    int n = c & (N_NODES - 1);
    if (fwd) { s = t;         d = n + T_TOK; }
    else     { s = n + T_TOK; d = t; }
    int sp = g_spd[(long)b * N_NODES + n];
    sp = sp < 0 ? 0 : (sp > SPD_L ? SPD_L : sp);
    w = ewp[sp];
  }
  out_src[idx] = s + off;
  out_dst[idx] = d + off;
  out_w[idx]   = w;
}

// --------------------------------- root ------------------------------------
__global__ void root_kernel(const int* __restrict__ g_root, int* __restrict__ out) {
  int i = threadIdx.x;
  if (i < BATCH) out[i] = g_root[i] + T_TOK + i * TPN;
}

// ------------------------------- launcher ----------------------------------
extern "C" void kernel_launch(void* const* d_in, const int* in_sizes, int n_in,
                              void* d_out, int out_size, void* d_ws, size_t ws_size,
                              hipStream_t stream) {
  const float* g_x    = (const float*)d_in[0];
  const float* g_pe   = (const float*)d_in[1];
  const int*   g_ei   = (const int*)  d_in[2];
  const int*   g_spd  = (const int*)  d_in[3];
  const int*   g_root = (const int*)  d_in[4];
  const float* tok    = (const float*)d_in[5];
  const float* tokpe  = (const float*)d_in[6];
  const float* sew    = (const float*)d_in[7];
  const float* ewp    = (const float*)d_in[8];

  float* out     = (float*)d_out;
  float* out_x   = out + OFF_X;
  float* out_pe  = out + OFF_PE;
  int*   out_src = (int*)(out + OFF_SRC);
  int*   out_dst = (int*)(out + OFF_DST);
  float* out_w   = out + OFF_W;
  int*   out_rt  = (int*)(out + OFF_ROOT);

  // x: 33,636,352 floats = 8,409,088 float4 = 8212 * (256*4)   (exact)
  x_copy_async<<<8212, 256, 0, stream>>>((const float4*)tok, (const float4*)g_x,
                                         (float4*)out_x);
  // pe: 8,409,088 floats = 2,102,272 float4 = 2053 * (256*4)   (exact)
  pe_copy_async<<<2053, 256, 0, stream>>>((const float4*)tokpe, (const float4*)g_pe,
                                          (float4*)out_pe);
  // edges: 9,442,944 threads
  long nblk = (NEDGE + 255) / 256;
  edges_kernel<<<(int)nblk, 256, 0, stream>>>(g_ei, g_spd, sew, ewp,
                                              out_src, out_dst, out_w);
  root_kernel<<<1, 64, 0, stream>>>(g_root, out_rt);
}